// ConvAttention_26929444946386
// MI455X (gfx1250) — compile-verified
//
#include <hip/hip_runtime.h>
#include <hip/hip_bf16.h>

typedef __attribute__((ext_vector_type(2))) float v2f;
typedef __attribute__((ext_vector_type(8))) float v8f;

#define BB   4
#define CC   128
#define HH   56
#define WW   56
#define NN   (HH * WW)        // 3136
#define NH   4
#define DD   32
#define QKVC (3 * CC)         // 384

__device__ __forceinline__ v8f wmma_f32(v2f a, v2f b, v8f c) {
  return __builtin_amdgcn_wmma_f32_16x16x4_f32(false, a, false, b, (short)0, c, false, false);
}

__device__ __forceinline__ int keyN(int ky0, int kx0, int j) {
  int y = ky0 + (j >> 3);
  int x = kx0 + (j & 7);
  if ((unsigned)y >= (unsigned)HH || (unsigned)x >= (unsigned)WW) return 0; // clamped; masked in softmax
  return y * WW + x;
}

// ---------------- Kernel 1: qkv[b][n][o] = sum_c x[b][c][n] * w_qkv[o][c] ----------------
// tile: 16 rows of n  x 16 cols of o, K=128 in steps of 4. One wave per tile.
__global__ __launch_bounds__(256) void qkv_gemm(const float* __restrict__ x,
                                                const float* __restrict__ wq,
                                                float* __restrict__ qkv) {
  const int wave = blockIdx.x * (blockDim.x >> 5) + (threadIdx.x >> 5);
  const int lane = threadIdx.x & 31;
  const int half = lane >> 4, l16 = lane & 15;

  const int MT = NN / 16, OT = QKVC / 16;          // 196, 24
  int t = wave;
  const int b  = t / (MT * OT); t -= b * MT * OT;
  const int mt = t / OT;
  const int ot = t - mt * OT;
  const int n0 = mt * 16, o0 = ot * 16;

  const float* xb = x + (size_t)b * CC * NN;
  v8f acc = {};
#pragma unroll 4
  for (int k0 = 0; k0 < CC; k0 += 4) {
    const int ka = k0 + 2 * half;
    v2f a, bfrag;
    a.x = xb[(size_t)ka * NN + n0 + l16];          // A[m=l16][ka]   (lane-contiguous)
    a.y = xb[(size_t)(ka + 1) * NN + n0 + l16];
    const float* wr = wq + (size_t)(o0 + l16) * CC + ka;
    bfrag.x = wr[0];                               // B[ka][n=l16] = w_qkv[o0+l16][ka]
    bfrag.y = wr[1];
    acc = wmma_f32(a, bfrag, acc);
  }
  float* out = qkv + ((size_t)b * NN + n0) * QKVC + o0 + l16;
#pragma unroll
  for (int r = 0; r < 8; ++r)
    out[(size_t)(r + 8 * half) * QKVC] = acc[r];   // D[r+8h][l16], lane-contiguous store
}

// ---------------- Kernel 2: windowed attention, one wave per (b, head, 4x4 query block) ----
__global__ __launch_bounds__(256) void local_attn(const float* __restrict__ qkv,
                                                  float* __restrict__ aout) {
  const int wv   = threadIdx.x >> 5;
  const int wave = blockIdx.x * 8 + wv;
  const int lane = threadIdx.x & 31;
  const int half = lane >> 4, l16 = lane & 15;

  int t = wave;
  const int b  = t / (NH * 14 * 14); t -= b * NH * 14 * 14;
  const int h  = t / (14 * 14);      t -= h * (14 * 14);
  const int by = t / 14;
  const int bx = t - by * 14;
  const int qy0 = by * 4, qx0 = bx * 4;
  const int ky0 = qy0 - 2, kx0 = qx0 - 2;          // 8x8 key superset

  __shared__ float smem_all[8 * 1040];             // per-wave: 16x64 scores + 16 rinv
  float* smem = smem_all + wv * 1040;

  const float* base = qkv + (size_t)b * NN * QKVC + h * DD;
  const float scale = 0.17677669529663688f;        // 32^-0.5

  // ---- scores S = (Q*scale) K^T : 16x64 in four 16x16 accumulators ----
  v8f acc[4] = {{}, {}, {}, {}};
  const int qn = (qy0 + (l16 >> 2)) * WW + qx0 + (l16 & 3);   // query row m=l16
#pragma unroll
  for (int k0 = 0; k0 < DD; k0 += 4) {
    const int ka = k0 + 2 * half;
    v2f a;
    a.x = base[(size_t)qn * QKVC + ka] * scale;
    a.y = base[(size_t)qn * QKVC + ka + 1] * scale;
#pragma unroll
    for (int tt = 0; tt < 4; ++tt) {
      const int kn = keyN(ky0, kx0, tt * 16 + l16);
      v2f bfrag;                                   // B[ka][j=l16] = K[key j][ka]
      bfrag.x = base[(size_t)kn * QKVC + CC + ka];
      bfrag.y = base[(size_t)kn * QKVC + CC + ka + 1];
      acc[tt] = wmma_f32(a, bfrag, acc[tt]);
    }
  }
  // dump D-layout scores to LDS row-major [16][64]
#pragma unroll
  for (int tt = 0; tt < 4; ++tt)
#pragma unroll
    for (int r = 0; r < 8; ++r)
      smem[(r + 8 * half) * 64 + tt * 16 + l16] = acc[tt][r];
  asm volatile("s_wait_dscnt 0" ::: "memory");

  // ---- masked softmax: lane m (<16) owns row m ----
  if (lane < 16) {
    const int m  = lane;
    const int qy = qy0 + (m >> 2), qx = qx0 + (m & 3);
    float mx = -3.0e38f;
    for (int j = 0; j < 64; ++j) {
      const int ky = ky0 + (j >> 3), kx = kx0 + (j & 7);
      const bool ok = (unsigned)ky < (unsigned)HH && (unsigned)kx < (unsigned)WW &&
                      (ky >= qy - 2) && (ky <= qy + 2) && (kx >= qx - 2) && (kx <= qx + 2);
      if (ok) mx = fmaxf(mx, smem[m * 64 + j]);
    }
    float sum = 0.f;
    for (int j = 0; j < 64; ++j) {
      const int ky = ky0 + (j >> 3), kx = kx0 + (j & 7);
      const bool ok = (unsigned)ky < (unsigned)HH && (unsigned)kx < (unsigned)WW &&
                      (ky >= qy - 2) && (ky <= qy + 2) && (kx >= qx - 2) && (kx <= qx + 2);
      const float e = ok ? __expf(smem[m * 64 + j] - mx) : 0.f;
      smem[m * 64 + j] = e;
      sum += e;
    }
    smem[1024 + m] = 1.0f / sum;                   // fold normalization into output store
  }
  asm volatile("s_wait_dscnt 0" ::: "memory");

  // ---- O = P V : 16x64 @ 64x32, A-fragments re-read from LDS ----
  v8f oacc[2] = {{}, {}};
#pragma unroll 4
  for (int j0 = 0; j0 < 64; j0 += 4) {
    const int ja = j0 + 2 * half;
    v2f a;
    a.x = smem[l16 * 64 + ja];                     // A[m=l16][ja]
    a.y = smem[l16 * 64 + ja + 1];
    const int kn0 = keyN(ky0, kx0, ja);
    const int kn1 = keyN(ky0, kx0, ja + 1);
#pragma unroll
    for (int tt = 0; tt < 2; ++tt) {
      v2f bfrag;                                   // B[ja][d=l16] = V[key ja][tt*16+l16]
      bfrag.x = base[(size_t)kn0 * QKVC + 2 * CC + tt * 16 + l16];
      bfrag.y = base[(size_t)kn1 * QKVC + 2 * CC + tt * 16 + l16];
      oacc[tt] = wmma_f32(a, bfrag, oacc[tt]);
    }
  }
#pragma unroll
  for (int r = 0; r < 8; ++r) {
    const int m = r + 8 * half;
    const float rinv = smem[1024 + m];
    const int qnm = (qy0 + (m >> 2)) * WW + qx0 + (m & 3);
    float* op = aout + ((size_t)b * NN + qnm) * CC + h * DD;
    op[l16]      = oacc[0][r] * rinv;
    op[16 + l16] = oacc[1][r] * rinv;
  }
}

// ---------------- Kernel 3: y[b][o][n] = sum_c aout[b][n][c]*w_proj[o][c] + b_proj[o] ------
// M = o (16), N = n (16) so the final [B,C,H,W] stores are lane-contiguous.
__global__ __launch_bounds__(256) void proj_gemm(const float* __restrict__ aout,
                                                 const float* __restrict__ wp,
                                                 const float* __restrict__ bp,
                                                 float* __restrict__ y) {
  const int wave = blockIdx.x * (blockDim.x >> 5) + (threadIdx.x >> 5);
  const int lane = threadIdx.x & 31;
  const int half = lane >> 4, l16 = lane & 15;

  const int OT = CC / 16, MT = NN / 16;            // 8, 196
  int t = wave;
  const int b  = t / (OT * MT); t -= b * OT * MT;
  const int ot = t / MT;
  const int nt = t - ot * MT;
  const int o0 = ot * 16, n0 = nt * 16;

  v8f acc = {};
#pragma unroll 4
  for (int k0 = 0; k0 < CC; k0 += 4) {
    const int ka = k0 + 2 * half;
    v2f a, bfrag;
    const float* wr = wp + (size_t)(o0 + l16) * CC + ka;
    a.x = wr[0];                                   // A[m=o][k]
    a.y = wr[1];
    const float* ar = aout + ((size_t)b * NN + n0 + l16) * CC + ka;
    bfrag.x = ar[0];                               // B[k][n=l16] = aout[n][k]
    bfrag.y = ar[1];
    acc = wmma_f32(a, bfrag, acc);
  }
#pragma unroll
  for (int r = 0; r < 8; ++r) {
    const int o = o0 + r + 8 * half;
    y[(size_t)b * CC * NN + (size_t)o * NN + n0 + l16] = acc[r] + bp[o];
  }
}

extern "C" void kernel_launch(void* const* d_in, const int* in_sizes, int n_in,
                              void* d_out, int out_size, void* d_ws, size_t ws_size,
                              hipStream_t stream) {
  const float* x      = (const float*)d_in[0];
  const float* w_qkv  = (const float*)d_in[1];
  const float* w_proj = (const float*)d_in[2];
  const float* b_proj = (const float*)d_in[3];
  float* out = (float*)d_out;

  float* qkv  = (float*)d_ws;                      // [B][N][384]  ~19.3 MB
  float* aout = qkv + (size_t)BB * NN * QKVC;      // [B][N][128]  ~6.4 MB

  // 4*196*24 = 18816 waves -> 2352 blocks of 8 waves
  qkv_gemm<<<2352, 256, 0, stream>>>(x, w_qkv, qkv);
  // 4*4*14*14 = 3136 waves -> 392 blocks
  local_attn<<<392, 256, 0, stream>>>(qkv, aout);
  // 4*8*196 = 6272 waves -> 784 blocks
  proj_gemm<<<784, 256, 0, stream>>>(aout, w_proj, b_proj, out);
}